// FLAME_77979426226428
// MI455X (gfx1250) — compile-verified
//
#include <hip/hip_runtime.h>
#include <hip/hip_bf16.h>

// ---------------------------------------------------------------------------
// FLAME LBS for MI455X (gfx1250, wave32).
// Main GEMM: C(1024 x 15069) = [betas|pose_feat](1024x192) * Wcat(192 x 15069)
// via V_WMMA_F32_16X16X4_F32 (exact fp32), fused with LBS skinning through
// LDS so v_posed never touches HBM.
// Register/LDS blocking: each block = 64 batches x 16 vertices (48 cols);
// 3 waves x 4 M-tiles each; A tile staged in LDS once per block, W fragment
// reused across the 4 M accumulators -> ~3.5x less L2 traffic than naive.
// ---------------------------------------------------------------------------

typedef __attribute__((ext_vector_type(2))) float v2f;
typedef __attribute__((ext_vector_type(4))) float v4f;
typedef __attribute__((ext_vector_type(8))) float v8f;

#define V_NUM   5023
#define B_NUM   1024
#define NJ      5
#define NCOL    15069        // V*3
#define NCOL_P  15072        // padded to 48*314
#define KDIM    192          // 150 betas + 36 pose_feature + 6 zero pad
#define N_LMK   68
#define MBLK    64           // batches per block (4 M-tiles)

// ---- workspace layout (float offsets) ----
#define WS_A    0                                  // 1024*192
#define WS_W    (WS_A + B_NUM*KDIM)                // 15072*192
#define WS_JS   (WS_W + NCOL_P*KDIM)               // 15*151 (pad 2272)
#define WS_REL  (WS_JS + 2272)                     // 1024*60

// ---------------------------------------------------------------------------
// Pack Wcat[n][k]: k<150 -> shapedirs (K-contiguous), k in [150,186) ->
// posedirs row (k-150) column n. Zero padding elsewhere (incl. n >= 15069).
// ---------------------------------------------------------------------------
__global__ void pack_w_kernel(const float* __restrict__ sd,
                              const float* __restrict__ pd,
                              float* __restrict__ W) {
    int idx = blockIdx.x * 256 + threadIdx.x;
    if (idx >= NCOL_P * KDIM) return;
    int n = idx / KDIM;
    int k = idx - n * KDIM;
    float v = 0.0f;
    if (n < NCOL) {
        if (k < 150)        v = sd[n * 150 + k];
        else if (k < 186)   v = pd[(k - 150) * NCOL + n];
    }
    W[idx] = v;
}

// ---------------------------------------------------------------------------
// JS[jc][l] = sum_v Jreg[j][v] * shapedirs[v][c][l]   (l < 150)
// JS[jc][150] = sum_v Jreg[j][v] * v_template[v][c]   (J0 term)
// ---------------------------------------------------------------------------
__global__ void js_kernel(const float* __restrict__ jreg,
                          const float* __restrict__ sd,
                          const float* __restrict__ vt,
                          float* __restrict__ JS) {
    int jc = blockIdx.x;            // 0..14  -> j = jc/3, c = jc%3
    int j = jc / 3, c = jc % 3;
    int l = threadIdx.x;
    if (l > 150) return;
    float acc = 0.0f;
    for (int v = 0; v < V_NUM; ++v) {
        float w = jreg[j * V_NUM + v];
        float x = (l < 150) ? sd[(v * 3 + c) * 150 + l] : vt[v * 3 + c];
        acc += w * x;
    }
    JS[jc * 151 + l] = acc;
}

// ---------------------------------------------------------------------------
// Per-batch prep (1 thread per batch): A row, Rodrigues, pose_feature,
// J via JS, kinematic chain, rel_transforms (5 x 3x4) -> ws.
// ---------------------------------------------------------------------------
__global__ void batch_prep_kernel(const float* __restrict__ shp,
                                  const float* __restrict__ expr,
                                  const float* __restrict__ pose,
                                  const float* __restrict__ eye,
                                  const float* __restrict__ neck,
                                  const float* __restrict__ JS,
                                  float* __restrict__ A,
                                  float* __restrict__ rel) {
    int b = blockIdx.x * blockDim.x + threadIdx.x;
    if (b >= B_NUM) return;
    float* Ab = A + (size_t)b * KDIM;
    for (int i = 0; i < 100; ++i) Ab[i]       = shp[b * 100 + i];
    for (int i = 0; i < 50;  ++i) Ab[100 + i] = expr[b * 50 + i];
    for (int i = 186; i < 192; ++i) Ab[i] = 0.0f;

    // full_pose = [pose[:3], neck, pose[3:6], eye(6)]
    float fp[15];
    fp[0] = pose[b * 6 + 0]; fp[1] = pose[b * 6 + 1]; fp[2] = pose[b * 6 + 2];
    fp[3] = neck[0]; fp[4] = neck[1]; fp[5] = neck[2];
    fp[6] = pose[b * 6 + 3]; fp[7] = pose[b * 6 + 4]; fp[8] = pose[b * 6 + 5];
    for (int i = 0; i < 6; ++i) fp[9 + i] = eye[i];

    // Rodrigues, matching reference: angle = ||v + 1e-8||, dir = v/angle
    float R[NJ][9];
    #pragma unroll
    for (int j = 0; j < NJ; ++j) {
        float x = fp[3 * j], y = fp[3 * j + 1], z = fp[3 * j + 2];
        float ax = x + 1e-8f, ay = y + 1e-8f, az = z + 1e-8f;
        float ang = sqrtf(ax * ax + ay * ay + az * az);
        float inv = 1.0f / ang;
        float rx = x * inv, ry = y * inv, rz = z * inv;
        float s = sinf(ang), c = cosf(ang), o = 1.0f - c;
        R[j][0] = 1.0f + o * (-(ry * ry + rz * rz));
        R[j][1] = -s * rz + o * (rx * ry);
        R[j][2] =  s * ry + o * (rx * rz);
        R[j][3] =  s * rz + o * (rx * ry);
        R[j][4] = 1.0f + o * (-(rx * rx + rz * rz));
        R[j][5] = -s * rx + o * (ry * rz);
        R[j][6] = -s * ry + o * (rx * rz);
        R[j][7] =  s * rx + o * (ry * rz);
        R[j][8] = 1.0f + o * (-(rx * rx + ry * ry));
    }
    // pose_feature = (R[1:] - I) flattened, row-major per joint
    #pragma unroll
    for (int j = 1; j < NJ; ++j)
        #pragma unroll
        for (int e = 0; e < 9; ++e) {
            float id = (e == 0 || e == 4 || e == 8) ? 1.0f : 0.0f;
            Ab[150 + (j - 1) * 9 + e] = R[j][e] - id;
        }

    // J[j][c] = J0 + JS . betas
    float J[NJ][3];
    #pragma unroll
    for (int j = 0; j < NJ; ++j)
        #pragma unroll
        for (int c = 0; c < 3; ++c) {
            const float* row = JS + (j * 3 + c) * 151;
            float acc = row[150];
            for (int l = 0; l < 150; ++l) acc += row[l] * Ab[l];
            J[j][c] = acc;
        }

    // kinematic chain, parents = {-1,0,1,1,1}; T = 3x4 affine rows
    const int PAR[NJ] = {-1, 0, 1, 1, 1};
    float T[NJ][12];
    #pragma unroll
    for (int i = 0; i < 3; ++i) {
        T[0][i * 4 + 0] = R[0][i * 3 + 0];
        T[0][i * 4 + 1] = R[0][i * 3 + 1];
        T[0][i * 4 + 2] = R[0][i * 3 + 2];
        T[0][i * 4 + 3] = J[0][i];
    }
    #pragma unroll
    for (int j = 1; j < NJ; ++j) {
        int p = PAR[j];
        float rj[3];
        #pragma unroll
        for (int c = 0; c < 3; ++c) rj[c] = J[j][c] - J[p][c];
        #pragma unroll
        for (int i = 0; i < 3; ++i) {
            #pragma unroll
            for (int k = 0; k < 3; ++k) {
                float s = 0.0f;
                #pragma unroll
                for (int m = 0; m < 3; ++m) s += T[p][i * 4 + m] * R[j][m * 3 + k];
                T[j][i * 4 + k] = s;
            }
            float t = T[p][i * 4 + 3];
            #pragma unroll
            for (int m = 0; m < 3; ++m) t += T[p][i * 4 + m] * rj[m];
            T[j][i * 4 + 3] = t;
        }
    }
    // rel_transforms: translation -= R_chain * J[j]
    float* relb = rel + (size_t)b * 60;
    #pragma unroll
    for (int j = 0; j < NJ; ++j)
        #pragma unroll
        for (int i = 0; i < 3; ++i) {
            relb[j * 12 + i * 4 + 0] = T[j][i * 4 + 0];
            relb[j * 12 + i * 4 + 1] = T[j][i * 4 + 1];
            relb[j * 12 + i * 4 + 2] = T[j][i * 4 + 2];
            float t = T[j][i * 4 + 3];
            #pragma unroll
            for (int k = 0; k < 3; ++k) t -= T[j][i * 4 + k] * J[j][k];
            relb[j * 12 + i * 4 + 3] = t;
        }
}

// ---------------------------------------------------------------------------
// WMMA GEMM + fused skinning, 64-batch block.
// Block = 96 threads (3 waves). A tile (64x192, 49KB) staged in LDS once;
// wave wv streams W fragments for its 16 columns and accumulates 4 M-tiles,
// reusing each W fragment across 4 WMMAs. LDS is re-used (after a barrier)
// for the v_posed / rel_transforms / weights staging in the epilogue.
// Fragment layouts (ISA 7.12.2, wave32):
//   A 16x4 f32 : lane&15 = M, VGPR r holds K = 2*(lane>>4)+r
//   B 4x16 f32 : lane&15 = N, VGPR r holds K = 2*(lane>>4)+r
//   C 16x16 f32: VGPR i holds M = i + 8*(lane>>4), N = lane&15
// ---------------------------------------------------------------------------
#define SM_VPOS 0                       // 64*48 = 3072 floats
#define SM_REL  3072                    // 64*60 = 3840 floats
#define SM_W    6912                    // 16*5  = 80 floats
__global__ __launch_bounds__(96) void gemm_skin_kernel(
        const float* __restrict__ A, const float* __restrict__ W,
        const float* __restrict__ rel, const float* __restrict__ vt,
        const float* __restrict__ lw, float* __restrict__ out) {
    __shared__ float smem[MBLK * KDIM];         // 12288 floats = 48KB

    const int tid  = threadIdx.x;
    const int wv   = tid >> 5;          // wave 0..2 -> column sub-tile
    const int lane = tid & 31;
    const int l15  = lane & 15;
    const int kh   = lane >> 4;

    const int b0 = blockIdx.y * MBLK;   // batch block (64)
    const int v0 = blockIdx.x * 16;     // vertex tile
    const int ncol = v0 * 3 + wv * 16 + l15;   // global column (< 15072, padded)

    // ---- stage A tile (contiguous 64x192 floats) into LDS ----
    {
        const v4f* gA = (const v4f*)(A + (size_t)b0 * KDIM);
        v4f* sA = (v4f*)smem;
        #pragma unroll
        for (int i = 0; i < (MBLK * KDIM / 4) / 96; ++i)
            sA[tid + i * 96] = gA[tid + i * 96];
    }
    __syncthreads();

    const float* wrow = W + (size_t)ncol * KDIM + kh * 2;
    const int aoff = l15 * KDIM + kh * 2;       // per-lane A fragment base

    v8f acc0 = {}, acc1 = {}, acc2 = {}, acc3 = {};
    #pragma unroll 4
    for (int k = 0; k < KDIM; k += 4) {
        v2f bv = *(const v2f*)(wrow + k);
        v2f a0 = *(const v2f*)(smem + aoff + k);
        v2f a1 = *(const v2f*)(smem + 16 * KDIM + aoff + k);
        v2f a2 = *(const v2f*)(smem + 32 * KDIM + aoff + k);
        v2f a3 = *(const v2f*)(smem + 48 * KDIM + aoff + k);
        acc0 = __builtin_amdgcn_wmma_f32_16x16x4_f32(false, a0, false, bv, (short)0, acc0, false, false);
        acc1 = __builtin_amdgcn_wmma_f32_16x16x4_f32(false, a1, false, bv, (short)0, acc1, false, false);
        acc2 = __builtin_amdgcn_wmma_f32_16x16x4_f32(false, a2, false, bv, (short)0, acc2, false, false);
        acc3 = __builtin_amdgcn_wmma_f32_16x16x4_f32(false, a3, false, bv, (short)0, acc3, false, false);
    }

    // ---- epilogue: re-use LDS (all A reads done) ----
    __syncthreads();

    float vtv = (ncol < NCOL) ? vt[ncol] : 0.0f;
    {
        const int col = wv * 16 + l15;
        #pragma unroll
        for (int i = 0; i < 8; ++i) {
            int m = kh * 8 + i;
            smem[SM_VPOS + (m +  0) * 48 + col] = acc0[i] + vtv;
            smem[SM_VPOS + (m + 16) * 48 + col] = acc1[i] + vtv;
            smem[SM_VPOS + (m + 32) * 48 + col] = acc2[i] + vtv;
            smem[SM_VPOS + (m + 48) * 48 + col] = acc3[i] + vtv;
        }
    }
    // stage rel_transforms (64 batches x 60) and lbs weights (16 verts x 5)
    for (int t = tid; t < MBLK * 60; t += 96) {
        int bl = t / 60, e = t - bl * 60;
        smem[SM_REL + bl * 60 + e] = rel[(size_t)(b0 + bl) * 60 + e];
    }
    if (tid < 80) {
        int vl = tid / 5, j = tid - vl * 5;
        int v = v0 + vl;
        smem[SM_W + vl * 5 + j] = (v < V_NUM) ? lw[v * 5 + j] : 0.0f;
    }
    __syncthreads();

    // 1024 skinning tasks (64 batches x 16 vertices)
    for (int task = tid; task < MBLK * 16; task += 96) {
        int bl = task >> 4, vl = task & 15;
        int v = v0 + vl;
        if (v >= V_NUM) continue;
        float Tm[12];
        #pragma unroll
        for (int e = 0; e < 12; ++e) {
            float s = 0.0f;
            #pragma unroll
            for (int j = 0; j < NJ; ++j)
                s += smem[SM_W + vl * 5 + j] * smem[SM_REL + bl * 60 + j * 12 + e];
            Tm[e] = s;
        }
        float px = smem[SM_VPOS + bl * 48 + vl * 3 + 0];
        float py = smem[SM_VPOS + bl * 48 + vl * 3 + 1];
        float pz = smem[SM_VPOS + bl * 48 + vl * 3 + 2];
        size_t o = ((size_t)(b0 + bl) * V_NUM + v) * 3;
        out[o + 0] = Tm[0] * px + Tm[1] * py + Tm[2]  * pz + Tm[3];
        out[o + 1] = Tm[4] * px + Tm[5] * py + Tm[6]  * pz + Tm[7];
        out[o + 2] = Tm[8] * px + Tm[9] * py + Tm[10] * pz + Tm[11];
    }
}

// ---------------------------------------------------------------------------
// Landmark gather: out[B,68,3] = verts[:, land_index, :]
// ---------------------------------------------------------------------------
__global__ void lmk_kernel(const int* __restrict__ li,
                           const float* __restrict__ verts,
                           float* __restrict__ out) {
    int idx = blockIdx.x * 256 + threadIdx.x;
    if (idx >= B_NUM * N_LMK * 3) return;
    int c = idx % 3;
    int r = idx / 3;
    int l = r % N_LMK;
    int b = r / N_LMK;
    int v = li[l];
    out[idx] = verts[((size_t)b * V_NUM + v) * 3 + c];
}

extern "C" void kernel_launch(void* const* d_in, const int* in_sizes, int n_in,
                              void* d_out, int out_size, void* d_ws, size_t ws_size,
                              hipStream_t stream) {
    const float* shp  = (const float*)d_in[0];   // (1024,100)
    const float* expr = (const float*)d_in[1];   // (1024,50)
    const float* pose = (const float*)d_in[2];   // (1024,6)
    const int*   li   = (const int*)  d_in[3];   // (68,)
    const float* vt   = (const float*)d_in[4];   // (5023,3)
    const float* sd   = (const float*)d_in[5];   // (5023,3,150)
    const float* pd   = (const float*)d_in[6];   // (36,15069)
    const float* jreg = (const float*)d_in[7];   // (5,5023)
    const float* lw   = (const float*)d_in[8];   // (5023,5)
    const float* eye  = (const float*)d_in[9];   // (6,)
    const float* neck = (const float*)d_in[10];  // (3,)

    float* ws   = (float*)d_ws;
    float* wsA  = ws + WS_A;
    float* wsW  = ws + WS_W;
    float* wsJS = ws + WS_JS;
    float* wsR  = ws + WS_REL;
    float* vout = (float*)d_out;                 // verts then landmarks
    float* lout = vout + (size_t)B_NUM * V_NUM * 3;

    pack_w_kernel<<<(NCOL_P * KDIM + 255) / 256, 256, 0, stream>>>(sd, pd, wsW);
    js_kernel<<<15, 160, 0, stream>>>(jreg, sd, vt, wsJS);
    batch_prep_kernel<<<B_NUM / 256, 256, 0, stream>>>(shp, expr, pose, eye, neck,
                                                       wsJS, wsA, wsR);
    dim3 grid((V_NUM + 15) / 16, B_NUM / MBLK); // 314 x 16
    gemm_skin_kernel<<<grid, 96, 0, stream>>>(wsA, wsW, wsR, vt, lw, vout);
    lmk_kernel<<<(B_NUM * N_LMK * 3 + 255) / 256, 256, 0, stream>>>(li, vout, lout);
}